// KGEModel_59734405152886
// MI455X (gfx1250) — compile-verified
//
#include <hip/hip_runtime.h>
#include <math.h>

// ---------------- problem constants (match reference) ----------------
#define M_TRIPLES 8192
#define N_REL     64
#define E_DIM     128
#define R_DIM     128
#define REG_LAMBDA 1e-5f
#define TILE_ROWS 16
#define MAX_TILES (M_TRIPLES / TILE_ROWS + N_REL)   // 512 + 64 = 576

typedef __attribute__((ext_vector_type(2))) float v2f;
typedef __attribute__((ext_vector_type(8))) float v8f;

// ---------------- workspace layout helpers ----------------
struct Ws {
    int*   counts;     // [64]
    int*   cursor;     // [64]
    int*   perm;       // [M]
    int*   tile_rel;   // [MAX_TILES]
    int*   tile_start; // [MAX_TILES]
    int*   tile_rows;  // [MAX_TILES]
    int*   ntiles;     // [1]
    float* loss_sum;   // [1]
    float* reg_sum;    // [1]
};

__host__ __device__ inline Ws ws_layout(void* base) {
    Ws w;
    int* p = (int*)base;
    w.counts     = p;             p += N_REL;
    w.cursor     = p;             p += N_REL;
    w.perm       = p;             p += M_TRIPLES;
    w.tile_rel   = p;             p += MAX_TILES;
    w.tile_start = p;             p += MAX_TILES;
    w.tile_rows  = p;             p += MAX_TILES;
    w.ntiles     = p;             p += 1;
    w.loss_sum   = (float*)p;     p += 1;
    w.reg_sum    = (float*)p;
    return w;
}

// ---------------- kernel 0: zero accumulators ----------------
__global__ void kg_init(int* counts, float* loss_sum, float* reg_sum) {
    int t = threadIdx.x;
    if (t < N_REL) counts[t] = 0;
    if (t == 0) { loss_sum[0] = 0.f; reg_sum[0] = 0.f; }
}

// ---------------- kernel 1: histogram of relations ----------------
__global__ void kg_count(const int* __restrict__ r, int* __restrict__ counts) {
    int m = blockIdx.x * blockDim.x + threadIdx.x;
    if (m < M_TRIPLES) atomicAdd(&counts[r[m]], 1);
}

// ---------------- kernel 2: prefix-sum + padded tile plan (1 thread) ----------------
__global__ void kg_plan(const int* __restrict__ counts, int* __restrict__ cursor,
                        int* __restrict__ tile_rel, int* __restrict__ tile_start,
                        int* __restrict__ tile_rows, int* __restrict__ ntiles) {
    int off = 0, nt = 0;
    for (int rel = 0; rel < N_REL; ++rel) {
        int c = counts[rel];
        cursor[rel] = off;
        for (int s = 0; s < c; s += TILE_ROWS) {
            tile_rel[nt]   = rel;
            tile_start[nt] = off + s;
            int rem = c - s;
            tile_rows[nt]  = rem < TILE_ROWS ? rem : TILE_ROWS;
            ++nt;
        }
        off += c;
    }
    *ntiles = nt;
}

// ---------------- kernel 3: scatter indices grouped by relation ----------------
__global__ void kg_scatter(const int* __restrict__ r, int* __restrict__ cursor,
                           int* __restrict__ perm) {
    int m = blockIdx.x * blockDim.x + threadIdx.x;
    if (m < M_TRIPLES) {
        int p = atomicAdd(&cursor[r[m]], 1);
        perm[p] = m;
    }
}

__device__ inline float dot4(float4 v) { return v.x*v.x + v.y*v.y + v.z*v.z + v.w*v.w; }

// ---------------- kernel 4: WMMA main ----------------
// One block (8 wave32) per 16-triple tile (single relation).
// Wave w computes output columns [16w, 16w+16) for all three projections.
__global__ __launch_bounds__(256)
void kg_main(const int* __restrict__ perm,
             const int* __restrict__ tile_rel,
             const int* __restrict__ tile_start,
             const int* __restrict__ tile_rows,
             const int* __restrict__ ntiles,
             const int* __restrict__ h_idx,
             const int* __restrict__ pos_idx,
             const int* __restrict__ neg_idx,
             const float* __restrict__ ent,
             const float* __restrict__ rel,
             const float* __restrict__ W,
             float* __restrict__ loss_sum,
             float* __restrict__ reg_sum)
{
    __shared__ float sh[TILE_ROWS * E_DIM];
    __shared__ float sp[TILE_ROWS * E_DIM];
    __shared__ float sn[TILE_ROWS * E_DIM];
    __shared__ float sr[R_DIM];
    __shared__ float s_pos[TILE_ROWS];
    __shared__ float s_neg[TILE_ROWS];
    __shared__ float s_reg;
    __shared__ int   trip[TILE_ROWS];

    const int tile = blockIdx.x;
    if (tile >= *ntiles) return;

    const int relid = tile_rel[tile];
    const int start = tile_start[tile];
    const int rows  = tile_rows[tile];

    const int t    = threadIdx.x;
    const int lane = t & 31;
    const int wave = t >> 5;

    if (t < TILE_ROWS) {
        s_pos[t] = 0.f;
        s_neg[t] = 0.f;
        trip[t]  = (t < rows) ? perm[start + t] : -1;
    }
    if (t == 0) s_reg = 0.f;
    __syncthreads();

    // ---- stage X tiles into LDS, fused L2-regularizer partial sums ----
    float regp = 0.f;
    // 16 rows * 32 float4 = 512 float4 per matrix; 2 per thread
    for (int i = t; i < TILE_ROWS * (E_DIM / 4); i += 256) {
        int row = i >> 5;
        int c4  = i & 31;
        int m   = trip[row];
        float4 vh = make_float4(0.f, 0.f, 0.f, 0.f), vp = vh, vn = vh;
        if (m >= 0) {
            vh = ((const float4*)(ent + (size_t)h_idx[m]   * E_DIM))[c4];
            vp = ((const float4*)(ent + (size_t)pos_idx[m] * E_DIM))[c4];
            vn = ((const float4*)(ent + (size_t)neg_idx[m] * E_DIM))[c4];
        }
        ((float4*)sh)[i] = vh;
        ((float4*)sp)[i] = vp;
        ((float4*)sn)[i] = vn;
        regp += dot4(vh) + dot4(vp) + dot4(vn);
    }
    // relation embedding (shared by all rows in the tile)
    if (t < R_DIM / 4) {
        float4 v = ((const float4*)(rel + (size_t)relid * R_DIM))[t];
        ((float4*)sr)[t] = v;
        regp += (float)rows * dot4(v);   // each valid row contributes l2(r_emb)
    }
    // block-reduce regularizer partials
    #pragma unroll
    for (int off = 16; off > 0; off >>= 1) regp += __shfl_xor(regp, off, 32);
    if (lane == 0) atomicAdd(&s_reg, regp);
    __syncthreads();
    if (t == 0) atomicAdd(reg_sum, 0.5f * s_reg);   // l2 = sum(x^2)/2

    // ---- WMMA: Y(16x128) = X(16x128) @ W(128x128), 16-col slice per wave ----
    const int n0   = wave * 16;
    const int half = lane >> 4;     // 0: lanes 0-15, 1: lanes 16-31
    const int lc   = lane & 15;
    const float* Wr = W + (size_t)relid * E_DIM * R_DIM;

    v8f ah = {}; v8f ap = {}; v8f an = {};

    for (int k = 0; k < E_DIM; k += 4) {
        // A fragment layout (16x4 f32): lanes 0-15 hold K=k,k+1; lanes 16-31 hold K=k+2,k+3
        const int ka = k + 2 * half;
        v2f fa_h = *(const v2f*)&sh[lc * E_DIM + ka];
        v2f fa_p = *(const v2f*)&sp[lc * E_DIM + ka];
        v2f fa_n = *(const v2f*)&sn[lc * E_DIM + ka];
        // B fragment (4x16 f32): VGPR0 = row K (lanes<16) / K+2 (lanes>=16), VGPR1 = next row
        v2f fb;
        fb[0] = Wr[(size_t)(ka)     * R_DIM + n0 + lc];
        fb[1] = Wr[(size_t)(ka + 1) * R_DIM + n0 + lc];
        if (k + 16 < E_DIM)
            __builtin_prefetch(&Wr[(size_t)(k + 16) * R_DIM + n0 + lc], 0, 1);

        ah = __builtin_amdgcn_wmma_f32_16x16x4_f32(false, fa_h, false, fb, (short)0, ah, false, false);
        ap = __builtin_amdgcn_wmma_f32_16x16x4_f32(false, fa_p, false, fb, (short)0, ap, false, false);
        an = __builtin_amdgcn_wmma_f32_16x16x4_f32(false, fa_n, false, fb, (short)0, an, false, false);
    }

    // ---- scores: C/D layout -> VGPR j holds rows j (lanes 0-15) and j+8 (lanes 16-31) ----
    const float re = sr[n0 + lc];
    #pragma unroll
    for (int j = 0; j < 8; ++j) {
        float hv = ah[j];
        float dp = hv + re - ap[j];
        float dn = hv + re - an[j];
        float dp2 = dp * dp;
        float dn2 = dn * dn;
        #pragma unroll
        for (int off = 8; off > 0; off >>= 1) {   // reduce within each 16-lane half
            dp2 += __shfl_xor(dp2, off, 32);
            dn2 += __shfl_xor(dn2, off, 32);
        }
        if (lc == 0) {
            int row = j + half * 8;
            atomicAdd(&s_pos[row], dp2);
            atomicAdd(&s_neg[row], dn2);
        }
    }
    __syncthreads();

    // ---- per-row loss: -log_sigmoid(pos - neg) = softplus(-(pos-neg)) ----
    if (t < TILE_ROWS && t < rows) {
        float x  = 0.5f * (s_pos[t] - s_neg[t]);   // pos_score - neg_score
        float lm = fmaxf(-x, 0.f) + log1pf(expf(-fabsf(x)));
        atomicAdd(loss_sum, lm);
    }
}

// ---------------- kernel 5: finalize scalar ----------------
__global__ void kg_final(const float* __restrict__ loss_sum,
                         const float* __restrict__ reg_sum,
                         float* __restrict__ out) {
    const float invM = 1.0f / (float)M_TRIPLES;
    out[0] = loss_sum[0] * invM + REG_LAMBDA * (reg_sum[0] * invM);
}

// ---------------- host launcher ----------------
extern "C" void kernel_launch(void* const* d_in, const int* in_sizes, int n_in,
                              void* d_out, int out_size, void* d_ws, size_t ws_size,
                              hipStream_t stream) {
    const int*   h       = (const int*)d_in[0];
    const int*   r       = (const int*)d_in[1];
    const int*   pos_t   = (const int*)d_in[2];
    const int*   neg_t   = (const int*)d_in[3];
    const float* ent     = (const float*)d_in[4];
    const float* rel     = (const float*)d_in[5];
    const float* W       = (const float*)d_in[6];
    float*       out     = (float*)d_out;

    Ws w = ws_layout(d_ws);

    kg_init<<<1, 256, 0, stream>>>(w.counts, w.loss_sum, w.reg_sum);
    kg_count<<<M_TRIPLES / 256, 256, 0, stream>>>(r, w.counts);
    kg_plan<<<1, 1, 0, stream>>>(w.counts, w.cursor, w.tile_rel, w.tile_start,
                                 w.tile_rows, w.ntiles);
    kg_scatter<<<M_TRIPLES / 256, 256, 0, stream>>>(r, w.cursor, w.perm);
    kg_main<<<MAX_TILES, 256, 0, stream>>>(w.perm, w.tile_rel, w.tile_start,
                                           w.tile_rows, w.ntiles,
                                           h, pos_t, neg_t, ent, rel, W,
                                           w.loss_sum, w.reg_sum);
    kg_final<<<1, 1, 0, stream>>>(w.loss_sum, w.reg_sum, out);
}